// AttentionHead_39496519254500
// MI455X (gfx1250) — compile-verified
//
#include <hip/hip_runtime.h>

typedef __attribute__((ext_vector_type(2))) float v2f;
typedef __attribute__((ext_vector_type(8))) float v8f;

// Problem constants (match reference)
constexpr int BB   = 8;
constexpr int TT   = 512;
constexpr int NV   = 25;
constexpr int CIN  = 128;
constexpr int DD   = 64;
constexpr int HEADS = BB * NV;          // 200
constexpr int NTOK  = BB * TT * NV;     // 102400

// LDS padding to kill bank conflicts (stride mod 64 dwords == 4)
constexpr int XPAD = 132;   // x tile rows (CIN=128 -> pad 132)
constexpr int KPAD = 68;    // K/V rows (DD=64 -> pad 68)
constexpr int PPAD = 17;    // P scratch rows (16 -> pad 17)

static __device__ __forceinline__ v8f wmma_f32(v2f a, v2f b, v8f c) {
    // V_WMMA_F32_16X16X4_F32: D = A(16x4) * B(4x16) + C, full fp32 precision.
    return __builtin_amdgcn_wmma_f32_16x16x4_f32(false, a, false, b, (short)0, c, false, false);
}

// ---------------------------------------------------------------------------
// Kernel 1: fused Q/K/V projection.
//   x: (NTOK, 128) row-major.  W*: (64,128).  Outputs per-head contiguous:
//   q_ws[head][t][e] (pre-scaled by 1/sqrt(64)), k_ws, v_ws likewise.
// Block: 256 threads (8 waves), 64 tokens per block, N = 192 fused outputs.
// ---------------------------------------------------------------------------
__global__ __launch_bounds__(256) void qkv_proj_kernel(
    const float* __restrict__ x,
    const float* __restrict__ Wq, const float* __restrict__ bq,
    const float* __restrict__ Wk, const float* __restrict__ bk,
    const float* __restrict__ Wv, const float* __restrict__ bv,
    float* __restrict__ q_ws, float* __restrict__ k_ws, float* __restrict__ v_ws)
{
    __shared__ __align__(16) float xs[64][XPAD];

    const int tid  = threadIdx.x;
    const int lane = tid & 31;
    const int wv   = tid >> 5;          // wave id 0..7
    const int hl   = (lane < 16) ? 0 : 1;
    const int ln   = lane & 15;
    const int m0   = blockIdx.x * 64;   // first token of this block

    // Stage 64x128 x-tile into LDS (coalesced float4).
    for (int idx = tid; idx < 64 * (CIN / 4); idx += 256) {
        const int row = idx >> 5;              // CIN/4 == 32
        const int c4  = (idx & 31) * 4;
        const float4 t4 = *(const float4*)(x + (size_t)(m0 + row) * CIN + c4);
        *(float4*)&xs[row][c4] = t4;
    }
    __syncthreads();

    // 4 M-tiles x 12 N-tiles = 48 WMMA output tiles; 6 per wave.
    for (int tl = wv; tl < 48; tl += 8) {
        const int tileM = tl & 3;
        const int tileN = tl >> 2;
        const int ng  = tileN * 16 + ln;       // 0..191 (this lane's N column)
        const int sel = ng >> 6;               // 0=q, 1=k, 2=v
        const int ni  = ng & 63;               // output channel within matrix
        const float* Wp = (sel == 0) ? Wq : (sel == 1) ? Wk : Wv;
        Wp += (size_t)ni * CIN;

        v8f acc = {};
        const int arow = tileM * 16 + ln;
        #pragma unroll
        for (int kk = 0; kk < CIN; kk += 4) {
            const int kb = kk + hl * 2;
            v2f a; a.x = xs[arow][kb]; a.y = xs[arow][kb + 1];
            const float2 bw = *(const float2*)(Wp + kb);
            v2f b; b.x = bw.x; b.y = bw.y;
            acc = wmma_f32(a, b, acc);
        }

        const float bias = ((sel == 0) ? bq : (sel == 1) ? bk : bv)[ni];
        float* outp = (sel == 0) ? q_ws : (sel == 1) ? k_ws : v_ws;
        const float qs = (sel == 0) ? 0.125f : 1.0f;  // fold 1/sqrt(64) into Q

        #pragma unroll
        for (int i = 0; i < 8; ++i) {
            const int m = m0 + tileM * 16 + i + hl * 8;   // token index
            const int vv = m % NV;
            const int t  = (m / NV) % TT;
            const int b  = m / (NV * TT);
            const int head = b * NV + vv;
            outp[((size_t)head * TT + t) * DD + ni] = (acc[i] + bias) * qs;
        }
    }
}

// ---------------------------------------------------------------------------
// Kernel 2: causal attention per head, flash-style online softmax.
// Reference subtlety: empty ring-buffer slots contribute logit 0 to the
// softmax denominator -> running max starts at 0 and (T-1-t)*exp(-m) is
// added to the denominator at the end.
// Grid: (HEADS, 2). Block: 256 threads (8 waves). K,V staged in LDS.
// ---------------------------------------------------------------------------
__global__ __launch_bounds__(256) void attn_kernel(
    const float* __restrict__ q_ws, const float* __restrict__ k_ws,
    const float* __restrict__ v_ws, float* __restrict__ out)
{
    extern __shared__ float lds[];
    float* Ksh = lds;                       // TT x KPAD
    float* Vsh = lds + TT * KPAD;           // TT x KPAD
    float* Psc = lds + 2 * TT * KPAD;       // 8 waves x 16 x PPAD

    const int head = blockIdx.x;
    const int half = blockIdx.y;            // splits Q chunks for parallelism
    const int tid  = threadIdx.x;
    const int lane = tid & 31;
    const int wv   = tid >> 5;
    const int hl   = (lane < 16) ? 0 : 1;
    const int ln   = lane & 15;

    const float* qg = q_ws + (size_t)head * TT * DD;
    const float* kg = k_ws + (size_t)head * TT * DD;
    const float* vg = v_ws + (size_t)head * TT * DD;

    // Stage K and V (512x64 each) into padded LDS.
    for (int idx = tid; idx < TT * (DD / 4); idx += 256) {
        const int row = idx >> 4;           // DD/4 == 16
        const int c4  = (idx & 15) * 4;
        const float4 k4 = *(const float4*)(kg + (size_t)row * DD + c4);
        const float4 v4 = *(const float4*)(vg + (size_t)row * DD + c4);
        *(float4*)&Ksh[row * KPAD + c4] = k4;
        *(float4*)&Vsh[row * KPAD + c4] = v4;
    }
    __syncthreads();

    float* Pw = Psc + wv * 16 * PPAD;       // per-wave scratch
    const int bidx = head / NV;
    const int vidx = head % NV;

    for (int c = 0; c < 2; ++c) {
        const int chunk = wv + 8 * half + 16 * c;   // 0..31
        const int t0 = chunk * 16;

        // Preload this wave's 16x64 Q chunk in WMMA A-layout.
        float2 qa[16];
        #pragma unroll
        for (int st = 0; st < 16; ++st) {
            const int kb = st * 4 + hl * 2;
            qa[st] = *(const float2*)(qg + (size_t)(t0 + ln) * DD + kb);
        }

        v8f acc[4] = {};                    // 16x64 output tile (4 N-tiles)
        float m[8], l[8];
        #pragma unroll
        for (int i = 0; i < 8; ++i) { m[i] = 0.0f; l[i] = 0.0f; }

        const int ntiles = t0 / 16 + 1;     // tiles up to & incl. diagonal
        for (int stile = 0; stile < ntiles; ++stile) {
            const int s0 = stile * 16;

            // S = Q * K^T  (16x16 tile), K-loop over Dk=64 in steps of 4.
            v8f s = {};
            #pragma unroll
            for (int st = 0; st < 16; ++st) {
                const int kb = st * 4 + hl * 2;
                const float2 bk2 = *(const float2*)&Ksh[(s0 + ln) * KPAD + kb];
                v2f a; a.x = qa[st].x; a.y = qa[st].y;
                v2f b; b.x = bk2.x;   b.y = bk2.y;
                s = wmma_f32(a, b, s);
            }

            // Online softmax update per row (row = i + 8*hl within chunk).
            const int sidx = s0 + ln;       // key position of this lane's col
            #pragma unroll
            for (int i = 0; i < 8; ++i) {
                const int t = t0 + i + hl * 8;
                const float val = s[i];
                const bool masked = (sidx > t);
                float mv = masked ? -3.0e38f : val;
                #pragma unroll
                for (int d = 1; d < 16; d <<= 1)
                    mv = fmaxf(mv, __shfl_xor(mv, d, 32));
                const float mnew = fmaxf(m[i], mv);
                const float sc = __expf(m[i] - mnew);
                float p = masked ? 0.0f : __expf(val - mnew);
                float rs = p;
                #pragma unroll
                for (int d = 1; d < 16; d <<= 1)
                    rs += __shfl_xor(rs, d, 32);
                l[i] = l[i] * sc + rs;
                m[i] = mnew;
                #pragma unroll
                for (int j = 0; j < 4; ++j) acc[j][i] *= sc;
                Pw[(i + hl * 8) * PPAD + ln] = p;   // C-layout -> LDS
            }
            // Wave-local LDS RAW fence before A-layout reload of P.
#if defined(__AMDGCN__)
            asm volatile("s_wait_dscnt 0" ::: "memory");
#endif

            // acc += P (16x16) * V_tile (16x64): 4 K-steps x 4 N-tiles.
            #pragma unroll
            for (int st = 0; st < 4; ++st) {
                const int kb = st * 4 + hl * 2;
                v2f pa;
                pa.x = Pw[ln * PPAD + kb];
                pa.y = Pw[ln * PPAD + kb + 1];
                #pragma unroll
                for (int j = 0; j < 4; ++j) {
                    const int n0 = j * 16;
                    v2f bvv;
                    bvv.x = Vsh[(s0 + kb)     * KPAD + n0 + ln];
                    bvv.y = Vsh[(s0 + kb + 1) * KPAD + n0 + ln];
                    acc[j] = wmma_f32(pa, bvv, acc[j]);
                }
            }
        }

        // Epilogue: add empty-slot denominator term, normalize, store.
        #pragma unroll
        for (int i = 0; i < 8; ++i) {
            const int t = t0 + i + hl * 8;
            const float denom = l[i] + (float)(TT - 1 - t) * __expf(-m[i]);
            const float inv = 1.0f / denom;
            const size_t base = ((size_t)(bidx * TT + t) * NV + vidx) * DD;
            #pragma unroll
            for (int j = 0; j < 4; ++j)
                out[base + j * 16 + ln] = acc[j][i] * inv;
        }
    }
}

// ---------------------------------------------------------------------------
extern "C" void kernel_launch(void* const* d_in, const int* in_sizes, int n_in,
                              void* d_out, int out_size, void* d_ws, size_t ws_size,
                              hipStream_t stream)
{
    const float* x  = (const float*)d_in[0];
    const float* Wq = (const float*)d_in[1];
    const float* bq = (const float*)d_in[2];
    const float* Wk = (const float*)d_in[3];
    const float* bk = (const float*)d_in[4];
    const float* Wv = (const float*)d_in[5];
    const float* bv = (const float*)d_in[6];
    float* out = (float*)d_out;

    // Workspace: q|k|v per-head contiguous, 3 * 200*512*64 floats = 78.6 MB.
    float* q_ws = (float*)d_ws;
    float* k_ws = q_ws + (size_t)HEADS * TT * DD;
    float* v_ws = k_ws + (size_t)HEADS * TT * DD;

    // Kernel 1: 102400 tokens / 64 per block.
    qkv_proj_kernel<<<NTOK / 64, 256, 0, stream>>>(
        x, Wq, bq, Wk, bk, Wv, bv, q_ws, k_ws, v_ws);

    // Kernel 2: K+V (padded) + per-wave P scratch in dynamic LDS.
    constexpr size_t SMEM2 =
        (size_t)(2 * TT * KPAD + 8 * 16 * PPAD) * sizeof(float);  // 287,232 B
    static_assert(SMEM2 <= 320u * 1024u, "exceeds WGP LDS");
    (void)hipFuncSetAttribute((const void*)attn_kernel,
                              hipFuncAttributeMaxDynamicSharedMemorySize,
                              (int)SMEM2);
    dim3 grid2(HEADS, 2);
    attn_kernel<<<grid2, 256, SMEM2, stream>>>(q_ws, k_ws, v_ws, out);
}